// CachedMemory_1348619731447
// MI455X (gfx1250) — compile-verified
//
#include <hip/hip_runtime.h>

typedef __attribute__((ext_vector_type(2))) float v2f;
typedef __attribute__((ext_vector_type(8))) float v8f;

#define DIMS 256
#define SPLITS 50
#define LDA 260            // LDS row stride (floats): 260%64=4 -> 16 rows hit distinct bank pairs
#define SIM_THRESHOLD 0.8f

// ---------------- reciprocal L2 norms: r = 1 / max(||row||, 1e-12) ----------------
__global__ void rnorm_kernel(const float* __restrict__ x, float* __restrict__ r, int rows) {
    int lane = threadIdx.x & 31;
    int wave = threadIdx.x >> 5;
    int row  = blockIdx.x * (blockDim.x >> 5) + wave;
    if (row >= rows) return;
    const float* p = x + (size_t)row * DIMS;
    float s = 0.f;
#pragma unroll
    for (int j = 0; j < DIMS / 32; ++j) { float v = p[lane + 32 * j]; s += v * v; }
#pragma unroll
    for (int m = 16; m >= 1; m >>= 1) s += __shfl_xor(s, m, 32);
    if (lane == 0) r[row] = 1.0f / fmaxf(sqrtf(s), 1e-12f);
}

// ---------------- copy original memory into outputs (keys f32, values -> float) ----------------
__global__ void copy_mem_kernel(const float* __restrict__ keys, const int* __restrict__ vals,
                                float* __restrict__ okeys, float* __restrict__ ovals, int M) {
    size_t nk = (size_t)M * DIMS;
    size_t total = nk + (size_t)M;
    for (size_t i = (size_t)blockIdx.x * blockDim.x + threadIdx.x; i < total;
         i += (size_t)gridDim.x * blockDim.x) {
        if (i < nk) okeys[i] = keys[i];
        else        ovals[i - nk] = (float)vals[i - nk];
    }
}

// ---------------- retrieve GEMM: per-(16-query block, key split) running max/argmax ----------------
__global__ __launch_bounds__(32)
void sim_argmax_kernel(const float* __restrict__ q, const float* __restrict__ keys,
                       const float* __restrict__ rq, const float* __restrict__ rn,
                       float* __restrict__ pmax, int* __restrict__ pidx, int M) {
    __shared__ float As[16 * LDA];
    __shared__ float sv[8][32];
    __shared__ int   si[8][32];

    const int lane  = threadIdx.x;
    const int qbase = blockIdx.x * 16;
    const int split = blockIdx.y;

    // Stage scaled A tile (16 queries x 256) into LDS.
    for (int i = lane; i < 16 * DIMS; i += 32) {
        int row = i >> 8, col = i & 255;
        As[row * LDA + col] = q[(size_t)(qbase + row) * DIMS + col] * rq[qbase + row];
    }
    __syncthreads();

    const int nTiles   = M >> 4;
    const int tilesPer = (nTiles + SPLITS - 1) / SPLITS;
    const int t0 = split * tilesPer;
    const int t1 = (t0 + tilesPer < nTiles) ? (t0 + tilesPer) : nTiles;

    const int nl   = lane & 15;           // key column of C owned by this lane
    const int koff = (lane >> 4) * 2;     // ISA A/B lane->K mapping for 16x16x4 f32

    const float INF = __builtin_inff();
    float bestv[8]; int besti[8];
#pragma unroll
    for (int r = 0; r < 8; ++r) { bestv[r] = -INF; besti[r] = 0; }

    for (int t = t0; t < t1; ++t) {
        const int nbase = t << 4;
        const float* brow = keys + (size_t)(nbase + nl) * DIMS;
        const float rnl = rn[nbase + nl];
        v8f c = {};
#pragma unroll 8
        for (int kk = 0; kk < DIMS; kk += 4) {
            v2f a = *(const v2f*)&As[nl * LDA + kk + koff];
            v2f b = *(const v2f*)&brow[kk + koff];
            c = __builtin_amdgcn_wmma_f32_16x16x4_f32(false, a, false, b,
                                                      (short)0, c, false, false);
        }
        const int idx = nbase + nl;
#pragma unroll
        for (int r = 0; r < 8; ++r) {
            float v = c[r] * rnl;   // fold key-side normalization into C (key fixed per lane)
            if (v > bestv[r] || (v == bestv[r] && idx < besti[r])) { bestv[r] = v; besti[r] = idx; }
        }
    }

#pragma unroll
    for (int r = 0; r < 8; ++r) { sv[r][lane] = bestv[r]; si[r][lane] = besti[r]; }
    __syncthreads();

    if (lane < 16) {
        const int qg  = lane;
        const int reg = qg & 7;
        const int lb  = (qg >> 3) * 16;   // C layout: M=reg (lanes0-15), M=reg+8 (lanes16-31)
        float bv = -INF; int bi = 0x7fffffff;
        for (int j = 0; j < 16; ++j) {
            float v = sv[reg][lb + j]; int ix = si[reg][lb + j];
            if (v > bv || (v == bv && ix < bi)) { bv = v; bi = ix; }
        }
        pmax[(size_t)(qbase + qg) * SPLITS + split] = bv;
        pidx[(size_t)(qbase + qg) * SPLITS + split] = bi;
    }
}

__global__ void retr_finalize_kernel(const float* __restrict__ pmax, const int* __restrict__ pidx,
                                     const int* __restrict__ memvals,
                                     float* __restrict__ out_retrieved, float* __restrict__ out_conf,
                                     int Q) {
    int qi = blockIdx.x * blockDim.x + threadIdx.x;
    if (qi >= Q) return;
    const float INF = __builtin_inff();
    float bv = -INF; int bi = 0x7fffffff;
    for (int s = 0; s < SPLITS; ++s) {
        float v = pmax[(size_t)qi * SPLITS + s]; int ix = pidx[(size_t)qi * SPLITS + s];
        if (v > bv || (v == bv && ix < bi)) { bv = v; bi = ix; }
    }
    out_retrieved[qi] = (float)memvals[bi];
    out_conf[qi] = bv;
}

// ---------------- update GEMM: max-sim, class-masked argmin, global argmin ----------------
__global__ __launch_bounds__(32)
void upd_sims_kernel(const float* __restrict__ u, const float* __restrict__ keys,
                     const float* __restrict__ ru, const float* __restrict__ rn,
                     const int* __restrict__ memvals, const int* __restrict__ labels,
                     float* __restrict__ pumax,
                     float* __restrict__ pmsV, int* __restrict__ pmsI,
                     float* __restrict__ pmaV, int* __restrict__ pmaI, int M) {
    __shared__ float As[16 * LDA];
    __shared__ int   lab[16];
    __shared__ float s0[8][32], s1[8][32], s2[8][32];
    __shared__ int   i1[8][32], i2[8][32];

    const int lane  = threadIdx.x;
    const int bbase = blockIdx.x * 16;
    const int split = blockIdx.y;

    for (int i = lane; i < 16 * DIMS; i += 32) {
        int row = i >> 8, col = i & 255;
        As[row * LDA + col] = u[(size_t)(bbase + row) * DIMS + col] * ru[bbase + row];
    }
    if (lane < 16) lab[lane] = labels[bbase + lane];
    __syncthreads();

    const int nTiles   = M >> 4;
    const int tilesPer = (nTiles + SPLITS - 1) / SPLITS;
    const int t0 = split * tilesPer;
    const int t1 = (t0 + tilesPer < nTiles) ? (t0 + tilesPer) : nTiles;

    const int nl   = lane & 15;
    const int koff = (lane >> 4) * 2;
    const int mhi  = (lane >> 4) * 8;   // C row offset for this lane half

    const float INF = __builtin_inff();
    float mx[8], msv[8], mav[8]; int msi[8], mai[8];
#pragma unroll
    for (int r = 0; r < 8; ++r) { mx[r] = -INF; msv[r] = INF; msi[r] = 0; mav[r] = INF; mai[r] = 0; }

    for (int t = t0; t < t1; ++t) {
        const int nbase = t << 4;
        const float* brow = keys + (size_t)(nbase + nl) * DIMS;
        const float rnl = rn[nbase + nl];
        const int   vn  = memvals[nbase + nl];
        v8f c = {};
#pragma unroll 8
        for (int kk = 0; kk < DIMS; kk += 4) {
            v2f a = *(const v2f*)&As[nl * LDA + kk + koff];
            v2f b = *(const v2f*)&brow[kk + koff];
            c = __builtin_amdgcn_wmma_f32_16x16x4_f32(false, a, false, b,
                                                      (short)0, c, false, false);
        }
        const int idx = nbase + nl;
#pragma unroll
        for (int r = 0; r < 8; ++r) {
            float v = c[r] * rnl;
            mx[r] = fmaxf(mx[r], v);
            if (vn == lab[r + mhi]) {
                if (v < msv[r] || (v == msv[r] && idx < msi[r])) { msv[r] = v; msi[r] = idx; }
            }
            if (v < mav[r] || (v == mav[r] && idx < mai[r])) { mav[r] = v; mai[r] = idx; }
        }
    }

#pragma unroll
    for (int r = 0; r < 8; ++r) {
        s0[r][lane] = mx[r];
        s1[r][lane] = msv[r]; i1[r][lane] = msi[r];
        s2[r][lane] = mav[r]; i2[r][lane] = mai[r];
    }
    __syncthreads();

    if (lane < 16) {
        const int bg  = lane;
        const int reg = bg & 7;
        const int lb  = (bg >> 3) * 16;
        float bmx = -INF, bsv = INF, bav = INF; int bsi = 0x7fffffff, bai = 0x7fffffff;
        for (int j = 0; j < 16; ++j) {
            bmx = fmaxf(bmx, s0[reg][lb + j]);
            float v = s1[reg][lb + j]; int ix = i1[reg][lb + j];
            if (v < bsv || (v == bsv && ix < bsi)) { bsv = v; bsi = ix; }
            v = s2[reg][lb + j]; ix = i2[reg][lb + j];
            if (v < bav || (v == bav && ix < bai)) { bav = v; bai = ix; }
        }
        size_t o = (size_t)(bbase + bg) * SPLITS + split;
        pumax[o] = bmx; pmsV[o] = bsv; pmsI[o] = bsi; pmaV[o] = bav; pmaI[o] = bai;
    }
}

__global__ void upd_finalize_kernel(const float* __restrict__ pumax,
                                    const float* __restrict__ pmsV, const int* __restrict__ pmsI,
                                    const float* __restrict__ pmaV, const int* __restrict__ pmaI,
                                    int* __restrict__ tgt, int* __restrict__ doupd, int B) {
    int b = blockIdx.x * blockDim.x + threadIdx.x;
    if (b >= B) return;
    const float INF = __builtin_inff();
    float mx = -INF, sv = INF, av = INF; int si = 0x7fffffff, ai = 0x7fffffff;
    for (int s = 0; s < SPLITS; ++s) {
        size_t o = (size_t)b * SPLITS + s;
        mx = fmaxf(mx, pumax[o]);
        float v = pmsV[o]; int ix = pmsI[o];
        if (v < sv || (v == sv && ix < si)) { sv = v; si = ix; }
        v = pmaV[o]; ix = pmaI[o];
        if (v < av || (v == av && ix < ai)) { av = v; ai = ix; }
    }
    doupd[b] = (mx < SIM_THRESHOLD) ? 1 : 0;
    tgt[b]   = (sv < INF) ? si : ai;   // has_same ? masked argmin : global argmin
}

// ---------------- sequential last-write-wins scatter (each column owned by one thread) ----------
__global__ __launch_bounds__(DIMS)
void scatter_kernel(const float* __restrict__ uf, const int* __restrict__ ulab,
                    const int* __restrict__ tgt, const int* __restrict__ doupd,
                    float* __restrict__ okeys, float* __restrict__ ovals, int B) {
    const int c = threadIdx.x;   // 0..255
    for (int b = 0; b < B; ++b) {
        if (doupd[b]) {
            int t = tgt[b];
            okeys[(size_t)t * DIMS + c] = uf[(size_t)b * DIMS + c];
            if (c == 0) ovals[t] = (float)ulab[b];
        }
    }
}

extern "C" void kernel_launch(void* const* d_in, const int* in_sizes, int n_in,
                              void* d_out, int out_size, void* d_ws, size_t ws_size,
                              hipStream_t stream) {
    const float* q     = (const float*)d_in[0];
    const float* keys  = (const float*)d_in[1];
    const int*   vals  = (const int*)d_in[2];
    const float* uf    = (const float*)d_in[3];
    const int*   ulab  = (const int*)d_in[4];

    const int Q = in_sizes[0] / DIMS;   // 2048
    const int M = in_sizes[2];          // 100000 (divisible by 16)
    const int B = in_sizes[4];          // 128

    float* out = (float*)d_out;
    float* out_retr = out;                         // [Q]
    float* out_conf = out + Q;                     // [Q]
    float* out_keys = out + 2 * (size_t)Q;         // [M*256]
    float* out_vals = out_keys + (size_t)M * DIMS; // [M]

    // workspace carve-up (all 4-byte entities)
    char* w = (char*)d_ws;
    float* rq    = (float*)w;              w += sizeof(float) * Q;
    float* rn    = (float*)w;              w += sizeof(float) * M;
    float* ru    = (float*)w;              w += sizeof(float) * B;
    float* pmax  = (float*)w;              w += sizeof(float) * (size_t)Q * SPLITS;
    int*   pidx  = (int*)w;                w += sizeof(int)   * (size_t)Q * SPLITS;
    float* pumax = (float*)w;              w += sizeof(float) * (size_t)B * SPLITS;
    float* pmsV  = (float*)w;              w += sizeof(float) * (size_t)B * SPLITS;
    int*   pmsI  = (int*)w;                w += sizeof(int)   * (size_t)B * SPLITS;
    float* pmaV  = (float*)w;              w += sizeof(float) * (size_t)B * SPLITS;
    int*   pmaI  = (int*)w;                w += sizeof(int)   * (size_t)B * SPLITS;
    int*   tgt   = (int*)w;                w += sizeof(int)   * B;
    int*   doupd = (int*)w;

    // 1) reciprocal norms (folded into GEMMs instead of materializing normalized copies)
    rnorm_kernel<<<(Q + 7) / 8, 256, 0, stream>>>(q, rq, Q);
    rnorm_kernel<<<(M + 7) / 8, 256, 0, stream>>>(keys, rn, M);
    rnorm_kernel<<<(B + 7) / 8, 256, 0, stream>>>(uf, ru, B);

    // 2) copy original memory to outputs (scatter patches it afterwards)
    copy_mem_kernel<<<4096, 256, 0, stream>>>(keys, vals, out_keys, out_vals, M);

    // 3) retrieve GEMM (f32 WMMA) + split-wise argmax, then finalize + label gather
    sim_argmax_kernel<<<dim3(Q / 16, SPLITS), 32, 0, stream>>>(q, keys, rq, rn, pmax, pidx, M);
    retr_finalize_kernel<<<(Q + 255) / 256, 256, 0, stream>>>(pmax, pidx, vals,
                                                              out_retr, out_conf, Q);

    // 4) update GEMM (f32 WMMA) + masked argmins, finalize, ordered scatter
    upd_sims_kernel<<<dim3(B / 16, SPLITS), 32, 0, stream>>>(uf, keys, ru, rn, vals, ulab,
                                                             pumax, pmsV, pmsI, pmaV, pmaI, M);
    upd_finalize_kernel<<<(B + 127) / 128, 128, 0, stream>>>(pumax, pmsV, pmsI, pmaV, pmaI,
                                                             tgt, doupd, B);
    scatter_kernel<<<1, DIMS, 0, stream>>>(uf, ulab, tgt, doupd, out_keys, out_vals, B);
}